// StuartLandauOscillator_6906307412148
// MI455X (gfx1250) — compile-verified
//
#include <hip/hip_runtime.h>
#include <math.h>

// Stuart-Landau oscillator bank, N = 4Mi, 32 Euler steps with global mean
// coupling. Bandwidth-bound FP32; state ping-pongs inside d_out (fits L2).
// Wave-level partial sums use V_WMMA_F32_16X16X4_F32 (A = ones): every row
// of D equals the 16 column sums, so D's VGPR0 + a 4-step ds_swizzle xor
// butterfly over the 16-group yields the full 32-lane total directly.

typedef __attribute__((ext_vector_type(2))) float v2f;
typedef __attribute__((ext_vector_type(8))) float v8f;

#define TPB    256
#define ITEMS  16
#define DT_C   0.01f
#define NSTEPS 32

// Group-of-32 xor swizzle: offset = {1'b0, xor[14:10], or[9:5]=0, and[4:0]=0x1F}
template <int PAT>
__device__ __forceinline__ float swz_add(float x) {
  const int y = __builtin_amdgcn_ds_swizzle(__float_as_int(x), PAT);
  return x + __int_as_float(y);
}

// Wave-wide sum of one f32 lane-value via WMMA.
// A = ones(16x4) => D[m][n] = colsum(n) for every m, so D VGPR0 lane l holds
// colsum(l%16); xor butterfly {1,2,4,8} then gives sum over all 16 columns =
// total over all 32 lanes of B, independent of B's (k,n)->lane packing.
// EXEC must be all ones at every call site (full blocks, uniform path).
__device__ __forceinline__ float wave_sum1(float a) {
  const v2f onesA = {1.0f, 1.0f};
  const v2f Ba = {a, 0.0f};
  const v8f cz = {};
  const v8f D = __builtin_amdgcn_wmma_f32_16x16x4_f32(false, onesA, false, Ba,
                                                      (short)0, cz, false,
                                                      false);
  float s = D[0];
  s = swz_add<0x041F>(s);  // xor 1
  s = swz_add<0x081F>(s);  // xor 2
  s = swz_add<0x101F>(s);  // xor 4
  s = swz_add<0x201F>(s);  // xor 8
  return s;
}

// Block-level deterministic pair-sum: WMMA wave sums -> LDS -> fixed-order
// fold by thread 0; writes the block partial pair to global memory.
__device__ __forceinline__ void block_sum2(float a, float b,
                                           float* __restrict__ dst2) {
  const float ta = wave_sum1(a);
  const float tb = wave_sum1(b);
  __shared__ float s[2 * (TPB / 32)];
  const int lane = threadIdx.x & 31;
  const int wv = threadIdx.x >> 5;
  if (lane == 0) { s[2 * wv] = ta; s[2 * wv + 1] = tb; }
  __syncthreads();
  if (threadIdx.x == 0) {
    float xr = 0.0f, xi = 0.0f;
#pragma unroll
    for (int w = 0; w < TPB / 32; ++w) { xr += s[2 * w]; xi += s[2 * w + 1]; }
    dst2[0] = xr;
    dst2[1] = xi;
  }
}

// ---------------------------------------------------------------------------
// Kernel: partial sums of the initial state (read-only d_in).
// ---------------------------------------------------------------------------
__global__ __launch_bounds__(TPB) void sl_reduce_init(
    const float* __restrict__ zr, const float* __restrict__ zi,
    float* __restrict__ parts, int n) {
  const int base = blockIdx.x * (TPB * ITEMS) + threadIdx.x;
  float ar = 0.0f, ai = 0.0f;
  if (base + (ITEMS - 1) * TPB < n) {  // block-uniform fast path
    float r[ITEMS], q[ITEMS];
#pragma unroll
    for (int k = 0; k < ITEMS; ++k) {
      r[k] = zr[base + k * TPB];
      q[k] = zi[base + k * TPB];
    }
#pragma unroll
    for (int k = 0; k < ITEMS; ++k) { ar += r[k]; ai += q[k]; }
  } else {
#pragma unroll
    for (int k = 0; k < ITEMS; ++k) {
      const int idx = base + k * TPB;
      if (idx < n) { ar += zr[idx]; ai += zi[idx]; }
    }
  }
  block_sum2(ar, ai, &parts[2 * blockIdx.x]);
}

// ---------------------------------------------------------------------------
// Kernel: fold block partial pairs -> mean (single block, fixed-order tree).
// ---------------------------------------------------------------------------
__global__ __launch_bounds__(TPB) void sl_finish_mean(
    const float* __restrict__ parts, int nPart, float* __restrict__ meanBuf,
    float fN) {
  __shared__ float sr[TPB];
  __shared__ float si[TPB];
  float ar = 0.0f, ai = 0.0f;
  for (int j = threadIdx.x; j < nPart; j += TPB) {
    ar += parts[2 * j];
    ai += parts[2 * j + 1];
  }
  sr[threadIdx.x] = ar;
  si[threadIdx.x] = ai;
  __syncthreads();
  for (int off = TPB / 2; off > 0; off >>= 1) {
    if (threadIdx.x < off) {
      sr[threadIdx.x] += sr[threadIdx.x + off];
      si[threadIdx.x] += si[threadIdx.x + off];
    }
    __syncthreads();
  }
  if (threadIdx.x == 0) {
    meanBuf[0] = sr[0] / fN;
    meanBuf[1] = si[0] / fN;
  }
}

// ---------------------------------------------------------------------------
// Kernel: one Euler step using the mean of the source state; also emits
// block partial sums of the NEW state for the next step's mean.
// ---------------------------------------------------------------------------
__global__ __launch_bounds__(TPB) void sl_step(
    const float* __restrict__ zr, const float* __restrict__ zi,
    const float* __restrict__ omega, const float* __restrict__ muP,
    const float* __restrict__ cplP, const float* __restrict__ meanBuf,
    float* __restrict__ ozr, float* __restrict__ ozi,
    float* __restrict__ parts, int n) {
  const float mu = muP[0];
  const float cpl = cplP[0];
  const float mr = meanBuf[0];
  const float mi = meanBuf[1];
  const int base = blockIdx.x * (TPB * ITEMS) + threadIdx.x;
  float ar = 0.0f, ai = 0.0f;
  if (base + (ITEMS - 1) * TPB < n) {  // block-uniform fast path
    float r[ITEMS], q[ITEMS], w[ITEMS];
#pragma unroll
    for (int k = 0; k < ITEMS; ++k) {
      r[k] = zr[base + k * TPB];
      q[k] = zi[base + k * TPB];
      w[k] = omega[base + k * TPB];
    }
#pragma unroll
    for (int k = 0; k < ITEMS; ++k) {
      const float a2 = r[k] * r[k] + q[k] * q[k];
      const float dzr = mu * r[k] - w[k] * q[k] - a2 * r[k] + cpl * (mr - r[k]);
      const float dzi = mu * q[k] + w[k] * r[k] - a2 * q[k] + cpl * (mi - q[k]);
      const float nr = r[k] + DT_C * dzr;
      const float ni = q[k] + DT_C * dzi;
      ozr[base + k * TPB] = nr;
      ozi[base + k * TPB] = ni;
      ar += nr;
      ai += ni;
    }
  } else {
#pragma unroll
    for (int k = 0; k < ITEMS; ++k) {
      const int idx = base + k * TPB;
      if (idx < n) {
        const float r = zr[idx];
        const float q = zi[idx];
        const float w = omega[idx];
        const float a2 = r * r + q * q;
        const float dzr = mu * r - w * q - a2 * r + cpl * (mr - r);
        const float dzi = mu * q + w * r - a2 * q + cpl * (mi - q);
        const float nr = r + DT_C * dzr;
        const float ni = q + DT_C * dzi;
        ozr[idx] = nr;
        ozi[idx] = ni;
        ar += nr;
        ai += ni;
      }
    }
  }
  block_sum2(ar, ai, &parts[2 * blockIdx.x]);
}

// ---------------------------------------------------------------------------
// Kernel: amplitudes + phases from the final state; block partial sums of
// amplitude for mean_amplitude.
// ---------------------------------------------------------------------------
__global__ __launch_bounds__(TPB) void sl_finalize(
    const float* __restrict__ zr, const float* __restrict__ zi,
    float* __restrict__ amp, float* __restrict__ phase,
    float* __restrict__ parts, int n) {
  const int base = blockIdx.x * (TPB * ITEMS) + threadIdx.x;
  float asum = 0.0f;
  if (base + (ITEMS - 1) * TPB < n) {  // block-uniform fast path
    float r[ITEMS], q[ITEMS];
#pragma unroll
    for (int k = 0; k < ITEMS; ++k) {
      r[k] = zr[base + k * TPB];
      q[k] = zi[base + k * TPB];
    }
#pragma unroll
    for (int k = 0; k < ITEMS; ++k) {
      const float a = sqrtf(r[k] * r[k] + q[k] * q[k]);
      amp[base + k * TPB] = a;
      phase[base + k * TPB] = atan2f(q[k], r[k]);
      asum += a;
    }
  } else {
#pragma unroll
    for (int k = 0; k < ITEMS; ++k) {
      const int idx = base + k * TPB;
      if (idx < n) {
        const float r = zr[idx];
        const float q = zi[idx];
        const float a = sqrtf(r * r + q * q);
        amp[idx] = a;
        phase[idx] = atan2f(q, r);
        asum += a;
      }
    }
  }
  const float ta = wave_sum1(asum);
  __shared__ float s[TPB / 32];
  const int lane = threadIdx.x & 31;
  const int wv = threadIdx.x >> 5;
  if (lane == 0) s[wv] = ta;
  __syncthreads();
  if (threadIdx.x == 0) {
    float x = 0.0f;
#pragma unroll
    for (int w = 0; w < TPB / 32; ++w) x += s[w];
    parts[blockIdx.x] = x;
  }
}

// ---------------------------------------------------------------------------
// Kernel: fold amplitude partials -> mean_amplitude; order parameter from
// the final mean (already in meanBuf after step 32's sl_finish_mean).
// ---------------------------------------------------------------------------
__global__ __launch_bounds__(TPB) void sl_scalars(
    const float* __restrict__ parts, int nPart,
    const float* __restrict__ meanBuf, float* __restrict__ orderP,
    float* __restrict__ meanAmp, float fN) {
  __shared__ float s[TPB];
  float a = 0.0f;
  for (int j = threadIdx.x; j < nPart; j += TPB) a += parts[j];
  s[threadIdx.x] = a;
  __syncthreads();
  for (int off = TPB / 2; off > 0; off >>= 1) {
    if (threadIdx.x < off) s[threadIdx.x] += s[threadIdx.x + off];
    __syncthreads();
  }
  if (threadIdx.x == 0) {
    *meanAmp = s[0] / fN;
    const float mr = meanBuf[0];
    const float mi = meanBuf[1];
    *orderP = sqrtf(mr * mr + mi * mi);
  }
}

// ---------------------------------------------------------------------------
extern "C" void kernel_launch(void* const* d_in, const int* in_sizes, int n_in,
                              void* d_out, int out_size, void* d_ws,
                              size_t ws_size, hipStream_t stream) {
  const float* zr0   = (const float*)d_in[0];
  const float* zi0   = (const float*)d_in[1];
  const float* omega = (const float*)d_in[2];
  const float* muP   = (const float*)d_in[3];
  const float* cplP  = (const float*)d_in[4];
  // d_in[5] = steps; fixed at 32 by the reference module constant.

  const int N = in_sizes[0];
  const float fN = (float)N;
  float* out = (float*)d_out;

  // Output layout (flat, return order):
  float* amp     = out;             // [0, N)
  float* phase   = out + N;         // [N, 2N)
  float* orderP  = out + 2 * N;     // scalar
  float* zrOut   = out + 2 * N + 1; // [2N+1, 3N+1)
  float* ziOut   = out + 3 * N + 1; // [3N+1, 4N+1)
  float* meanAmp = out + 4 * N + 1; // scalar

  // Ping-pong state buffers borrowed from d_out regions (all overwritten
  // later by the kernels that own those regions). Parity: step s (0-based)
  // even -> A (amp/phase slots), odd -> B (zr/zi slots); s=31 -> B, so the
  // final state lands exactly in the zr/zi output slots.
  float* Azr = amp;
  float* Azi = phase;
  float* Bzr = zrOut;
  float* Bzi = ziOut;

  const int blocks = (N + TPB * ITEMS - 1) / (TPB * ITEMS); // 1024 for 4Mi
  float* ws = (float*)d_ws;
  float* parts = ws;                 // 2*blocks floats (amp path uses blocks)
  float* meanBuf = ws + 2 * blocks;  // 2 floats

  // Mean of the initial state.
  sl_reduce_init<<<blocks, TPB, 0, stream>>>(zr0, zi0, parts, N);
  sl_finish_mean<<<1, TPB, 0, stream>>>(parts, blocks, meanBuf, fN);

  const float* sr = zr0;
  const float* si = zi0;
  for (int s = 0; s < NSTEPS; ++s) {
    float* dr = ((s & 1) == 0) ? Azr : Bzr;
    float* di = ((s & 1) == 0) ? Azi : Bzi;
    sl_step<<<blocks, TPB, 0, stream>>>(sr, si, omega, muP, cplP, meanBuf, dr,
                                        di, parts, N);
    sl_finish_mean<<<1, TPB, 0, stream>>>(parts, blocks, meanBuf, fN);
    sr = dr;
    si = di;
  }

  // Final state is in (Bzr, Bzi) == the zr/zi output slots.
  sl_finalize<<<blocks, TPB, 0, stream>>>(Bzr, Bzi, amp, phase, parts, N);
  sl_scalars<<<1, TPB, 0, stream>>>(parts, blocks, meanBuf, orderP, meanAmp,
                                    fN);
}